// RadianceField_2413771621037
// MI455X (gfx1250) — compile-verified
//
#include <hip/hip_runtime.h>
#include <stdint.h>

#define IDIM 128
#define NSAMP 128

typedef float v2f __attribute__((ext_vector_type(2)));
typedef float v4f __attribute__((ext_vector_type(4)));
typedef float v8f __attribute__((ext_vector_type(8)));

__device__ __forceinline__ int clampi(int v, int lo, int hi) {
    v = v < lo ? lo : v;
    return v > hi ? hi : v;
}

// One block (128 threads = 4 wave32) per ray.
__global__ __launch_bounds__(NSAMP) void rf_render_kernel(
    const float* __restrict__ xbuf, const float* __restrict__ dbuf,
    const float* __restrict__ ubuf, const float* __restrict__ grid,
    const float* __restrict__ opac, float* __restrict__ out)
{
    const int ray = blockIdx.x;
    const int tid = threadIdx.x;

    __shared__ float s_t[NSAMP];     // u -> t -> sorted t
    __shared__ float s_scan[NSAMP];  // dts -> chunk-local inclusive scan -> terms

    // ---- CDNA5 async global->LDS copy of u[ray, tid] into the sort buffer ----
    {
        const float* gp = ubuf + (size_t)ray * NSAMP + tid;
        unsigned lds_addr = (unsigned)(size_t)&s_t[tid];  // low 32 bits of flat addr = LDS offset
        asm volatile("global_load_async_to_lds_b32 %0, %1, off"
                     :: "v"(lds_addr), "v"(gp)
                     : "memory");
    }

    // ---- ray/AABB slab test (uniform across block; overlaps the async copy) ----
    const float ox = xbuf[ray * 3 + 0], oy = xbuf[ray * 3 + 1], oz = xbuf[ray * 3 + 2];
    const float dx = dbuf[ray * 3 + 0], dy = dbuf[ray * 3 + 1], dz = dbuf[ray * 3 + 2];
    const float INF = (float)IDIM * (float)IDIM * (float)IDIM;  // 2097152
    const float bmax = (float)(IDIM - 1);
    const float ivx = 1.0f / dx, ivy = 1.0f / dy, ivz = 1.0f / dz;
    const float t0x = (0.0f - ox) * ivx, t1x = (bmax - ox) * ivx;
    const float t0y = (0.0f - oy) * ivy, t1y = (bmax - oy) * ivy;
    const float t0z = (0.0f - oz) * ivz, t1z = (bmax - oz) * ivz;
    const float tmin = fmaxf(-INF, fmaxf(fmaxf(fminf(t0x, t1x), fminf(t0y, t1y)), fminf(t0z, t1z)));
    const float tmax = fminf( INF, fminf(fminf(fmaxf(t0x, t1x), fmaxf(t0y, t1y)), fmaxf(t0z, t1z)));
    const bool  mask   = tmin < tmax;
    const float tmin_s = mask ? tmin : 0.0f;
    const float tmax_s = mask ? tmax : 1.0f;

    asm volatile("s_wait_asynccnt 0" ::: "memory");
    __syncthreads();

    float t = tmin_s + s_t[tid] * (tmax_s - tmin_s);
    s_t[tid] = t;

    // ---- prefetch this (pre-sort) sample's voxel line: same address multiset
    //      as post-sort, warms L2/L1 while the bitonic sort runs ----
    {
        float px = ox + t * dx, py = oy + t * dy, pz = oz + t * dz;
        int i0 = clampi((int)floorf(px), 0, IDIM - 1);
        int i1 = clampi((int)floorf(py), 0, IDIM - 1);
        int i2 = clampi((int)floorf(pz), 0, IDIM - 1);
        size_t fl = ((size_t)i0 * IDIM + i1) * IDIM + i2;
        __builtin_prefetch(grid + fl * 9, 0, 1);   // -> global_prefetch_b8
        __builtin_prefetch(opac + fl, 0, 1);
    }

    // ---- bitonic sort of 128 sample positions (matches jnp.sort ascending) ----
    for (int k = 2; k <= NSAMP; k <<= 1) {
        for (int j = k >> 1; j > 0; j >>= 1) {
            __syncthreads();
            int ixj = tid ^ j;
            if (ixj > tid) {
                float a = s_t[tid], b = s_t[ixj];
                bool up = ((tid & k) == 0);
                if ((a > b) == up) { s_t[tid] = b; s_t[ixj] = a; }
            }
        }
    }
    __syncthreads();

    t = s_t[tid];
    const float tn = (tid < NSAMP - 1) ? s_t[tid + 1] : t;

    // ---- 8-corner trilinear + SH shading for this sample ----
    const float px = ox + t * dx, py = oy + t * dy, pz = oz + t * dz;
    const float f0 = floorf(px), f1 = floorf(py), f2 = floorf(pz);
    const float xd0 = px - f0, xd1 = py - f1, xd2 = pz - f2;

    float acc_h = 0.0f, acc_o = 0.0f;
#pragma unroll
    for (int n = 0; n < 8; ++n) {
        const float c0 = f0 + (float)(n & 1);         // DELTA_IJK ordering: bit0->x, bit1->y, bit2->z
        const float c1 = f1 + (float)((n >> 1) & 1);
        const float c2 = f2 + (float)((n >> 2) & 1);
        const int i0 = clampi((int)c0, 0, IDIM - 1);
        const int i1 = clampi((int)c1, 0, IDIM - 1);
        const int i2 = clampi((int)c2, 0, IDIM - 1);
        const size_t fl = ((size_t)i0 * IDIM + i1) * IDIM + i2;

        const float* gp = grid + fl * 9;
        v4f ga, gb;                               // 9 floats: b128 + b128 + b32
        __builtin_memcpy(&ga, gp, 16);
        __builtin_memcpy(&gb, gp + 4, 16);
        const float g8 = gp[8];
        const float op = opac[fl];

        // direction from ray origin to (unclipped) corner, normalized
        float vx = c0 - ox, vy = c1 - oy, vz = c2 - oz;
        const float rn = 1.0f / (sqrtf(vx * vx + vy * vy + vz * vz) + 1e-8f);
        vx *= rn; vy *= rn; vz *= rn;

        // sum_c K_c * basis_c * grid_c  (channel sum collapsed before weighting)
        const float sh = 0.28209479f * ga.x
                       + 0.48860251f * (vy * ga.y + vz * ga.z + vx * ga.w)
                       + 1.09254843f * (vx * vy * gb.x + vy * vz * gb.y + vx * vz * gb.w)
                       + 0.31539157f * (3.0f * vz * vz - 1.0f) * gb.z
                       + 0.54627422f * (vx * vx - vy * vy) * g8;

        const float w = ((n & 1) ? xd0 : 1.0f - xd0)
                      * ((n & 2) ? xd1 : 1.0f - xd1)
                      * ((n & 4) ? xd2 : 1.0f - xd2);
        acc_h += w * sh;
        acc_o += w * op;
    }

    const float color = 1.0f / (1.0f + expf(-acc_h));       // sigmoid(sum interp_harm)
    const float dts = (tid < NSAMP - 1) ? (tn - t) * acc_o : 0.0f;

    // ---- prefix sum of dts via WMMA: cumsum == lower-triangular-ones matmul ----
    // Dmat[i][c] = dts[16c+i] (16x8, cols 8..15 zero).  P = T16 x Dmat with
    // T16[m][k] = (k<=m), computed as 4 K-chained v_wmma_f32_16x16x4_f32.
    // P[i][c] = chunk-local inclusive prefix; carry fixed up from P[15][c'].
    s_scan[tid] = dts;
    __syncthreads();
    if (tid < 32) {                       // wave 0 only: EXEC all-ones inside
        const int col  = tid & 15;        // A's M, B/D's N
        const int half = tid >> 4;        // lane half selects K/M sub-block
        const bool valid = col < 8;       // real chunks are columns 0..7
        const int cbase = valid ? (col << 4) : 0;   // in-bounds even when padded
        v8f acc = {};
#pragma unroll
        for (int j = 0; j < 4; ++j) {
            const int k0 = 4 * j + 2 * half;        // K of VGPR r=0
            const int k1 = k0 + 1;                  // K of VGPR r=1
            v2f a, b;
            a.x = (k0 <= col) ? 1.0f : 0.0f;        // T16[m][k] = k<=m
            a.y = (k1 <= col) ? 1.0f : 0.0f;
            const float b0 = s_scan[cbase + k0];
            const float b1 = s_scan[cbase + k1];
            b.x = valid ? b0 : 0.0f;
            b.y = valid ? b1 : 0.0f;
            acc = __builtin_amdgcn_wmma_f32_16x16x4_f32(
                /*neg_a=*/false, a, /*neg_b=*/false, b,
                /*c_mod=*/(short)0, acc, /*reuse_a=*/false, /*reuse_b=*/false);
        }
        if (valid) {                      // D: lane l, VGPR r -> M=r+8*half, N=l%16
#pragma unroll
            for (int r = 0; r < 8; ++r)
                s_scan[(col << 4) + r + 8 * half] = acc[r];
        }
    }
    __syncthreads();

    const float incl = s_scan[tid];       // chunk-local inclusive prefix
    const int chunk = tid >> 4;
    float carry = 0.0f;
#pragma unroll
    for (int c = 0; c < 7; ++c)           // exclusive carry across chunk totals
        carry += (c < chunk) ? s_scan[(c << 4) + 15] : 0.0f;
    const float cum_excl = (incl + carry) - dts;   // inclusive -> exclusive

    const float term = (tid < NSAMP - 1)
        ? expf(-cum_excl) * (1.0f - expf(-dts)) * color
        : 0.0f;

    // ---- tree reduction of the compositing terms ----
    __syncthreads();
    s_scan[tid] = term;
#pragma unroll
    for (int off = NSAMP / 2; off > 0; off >>= 1) {
        __syncthreads();
        if (tid < off) s_scan[tid] += s_scan[tid + off];
    }
    __syncthreads();
    if (tid == 0) out[ray] = mask ? s_scan[0] : 0.0f;
}

extern "C" void kernel_launch(void* const* d_in, const int* in_sizes, int n_in,
                              void* d_out, int out_size, void* d_ws, size_t ws_size,
                              hipStream_t stream) {
    const float* x    = (const float*)d_in[0];  // [R,3]
    const float* d    = (const float*)d_in[1];  // [R,3]
    const float* u    = (const float*)d_in[2];  // [R,128]
    const float* grid = (const float*)d_in[3];  // [128^3, 9]
    const float* op   = (const float*)d_in[4];  // [128^3]
    float* out = (float*)d_out;                 // [R]

    const int nrays = in_sizes[0] / 3;
    rf_render_kernel<<<nrays, NSAMP, 0, stream>>>(x, d, u, grid, op, out);
}